// my_NGP_INTERP_ENC_7017976562279
// MI455X (gfx1250) — compile-verified
//
#include <hip/hip_runtime.h>
#include <math.h>

// NGP-style multi-level geodesic-weighted grid interpolation encoder.
// B points, 4 levels, 2 features/corner, bilinear blend, out = (B, 8) f32.
//
// Memory-bound design: ~113 MB total HBM traffic (~5us floor at 23.3 TB/s),
// gathers L2-resident (33 MB tables << 192 MB L2). 8x b128 gathers/point via
// explicit 16B row loads (align-8), saddr + 32-bit unsigned voffset
// addressing, 4-wide weight math (levels in components) for pk/VOPD packing,
// NT temporal hints on the streaming point load and output stores.

typedef __attribute__((ext_vector_type(2))) float vf2;
typedef __attribute__((ext_vector_type(4))) float vf4;

#define NGP_TABLE ((720 + 1) * (1440 + 1)) /* 1,038,961 entries per level */

__device__ __forceinline__ vf4 splat4(float v) { return (vf4){v, v, v, v}; }
__device__ __forceinline__ vf4 fma4(vf4 a, vf4 b, vf4 c) {
    return __builtin_elementwise_fma(a, b, c);
}

__global__ __launch_bounds__(256) void ngp_interp_enc_kernel(
    const float* __restrict__ x,     // (B, 2) lat, lon
    const float* __restrict__ emb,   // (4, NGP_TABLE, 2)
    float* __restrict__ out,         // (B, 8)
    int B)
{
    int i = blockIdx.x * 256 + (int)threadIdx.x;
    if (i >= B) return;

    // Streaming read of the query point (non-temporal: read once).
    vf2 p = __builtin_nontemporal_load(((const vf2*)x) + i);
    float lat = p.x;
    float lon = p.y;

    const float DEG2RAD = 0.017453292519943295f;
    const float BOX_LAT = -90.25f;   // LAT_MIN - FINEST
    const float BOX_LON = -0.25f;    // LON_MIN - FINEST

    // floor((a)/2^L) == floor(a) >> L exactly (a >= 0, /2^L exact in fp32),
    // so one fp->uint truncation per coordinate yields all level indices.
    unsigned ilat = (unsigned)(lat - BOX_LAT);   // [0.25, 180.25) -> trunc==floor
    unsigned ilon = (unsigned)(lon - BOX_LON);   // [0.25, 360.25)

    // ---- Issue all 8 corner-row gathers up front (one clause covers the full
    // latency window). Lon-adjacent corners are contiguous, so each row is one
    // 16B load (b128, align 8): row0 = {c00.f0, c00.f1, c01.f0, c01.f1},
    // row1 = {c10.f0, c10.f1, c11.f0, c11.f1}. Flat unsigned index (level
    // base folded in, < 2^31 bytes) keeps saddr + 32-bit voffset addressing.
    const int w_tab[4] = {1440, 720, 360, 180};
    const vf2* eb = (const vf2*)emb;
    vf4 row0[4], row1[4];
    unsigned blat_u[4], blon_u[4];
    #pragma unroll
    for (int L = 0; L < 4; ++L) {
        unsigned blat = ilat >> L;
        unsigned blon = ilon >> L;
        blat_u[L] = blat;
        blon_u[L] = blon;
        unsigned idx00 = (unsigned)(L * NGP_TABLE) + blat * (unsigned)w_tab[L] + blon;
        unsigned idx10 = idx00 + (unsigned)w_tab[L];
        __builtin_memcpy(&row0[L], eb + idx00, sizeof(vf4));
        __builtin_memcpy(&row1[L], eb + idx10, sizeof(vf4));
    }

    // ---- 4-wide weight math: component L = level L.
    const vf4 gsv  = {1.f, 2.f, 4.f, 8.f};
    const vf4 igsv = {1.f, 0.5f, 0.25f, 0.125f};
    // sin(gs * (pi/180) / 2) per level (denominator half-angle sine).
    const vf4 s2v  = {8.72653549837393e-3f, 1.74524064372835e-2f,
                      3.48994967025010e-2f, 6.97564737441253e-2f};

    vf4 blat_f = {(float)blat_u[0], (float)blat_u[1], (float)blat_u[2], (float)blat_u[3]};
    vf4 blon_f = {(float)blon_u[0], (float)blon_u[1], (float)blon_u[2], (float)blon_u[3]};
    vf4 gmin_lat = fma4(blat_f, gsv, splat4(BOX_LAT));
    vf4 gmin_lon = fma4(blon_f, gsv, splat4(BOX_LON));

    // wlat: meridian geodesic is linear in dlat (2*atan2(|sin d/2|,|cos d/2|)
    // == |d| for |d| <= pi), so the ratio is plain fractional position.
    vf4 wlat = (splat4(lat) - gmin_lat) * igsv;

    // wlon: parallel geodesic; 2*atan2(sqrt(a), sqrt(1-a)) == 2*asin(sqrt(a)),
    // sqrt(a) = |cos(phi)| * sin(dlon/2), dlon in (0, gs]. |phi| <= 1.576 rad
    // -> hardware v_cos_f32; sin/asin args <= 0.0699 -> short polynomials.
    vf4 phir = gmin_lat * splat4(DEG2RAD);
    vf4 cphi = {fabsf(__cosf(phir.x)), fabsf(__cosf(phir.y)),
                fabsf(__cosf(phir.z)), fabsf(__cosf(phir.w))};
    vf4 dlon = (gmin_lon + gsv) - splat4(lon);
    vf4 t  = dlon * splat4(0.5f * DEG2RAD);
    vf4 t2 = t * t;
    // sin(t) ~= t*(1 - t^2/6 + t^4/120), abs err ~1e-12
    vf4 s1 = t * fma4(t2, fma4(t2, splat4(8.3333333e-3f), splat4(-0.16666667f)),
                      splat4(1.f));
    vf4 yn  = cphi * s1;
    vf4 yd  = cphi * s2v;
    vf4 yn2 = yn * yn;
    vf4 yd2 = yd * yd;
    // asin(y) ~= y*(1 + y^2*(1/6 + y^2*(3/40))), rel err ~5e-9
    vf4 an = yn * fma4(yn2, fma4(yn2, splat4(0.075f), splat4(0.16666667f)), splat4(1.f));
    vf4 ad = yd * fma4(yd2, fma4(yd2, splat4(0.075f), splat4(0.16666667f)), splat4(1.f));
    vf4 rad = {__builtin_amdgcn_rcpf(ad.x), __builtin_amdgcn_rcpf(ad.y),
               __builtin_amdgcn_rcpf(ad.z), __builtin_amdgcn_rcpf(ad.w)};
    vf4 wlon = an * rad;

    // ---- Bilinear blend per level, directly on the row vectors (corner
    // order per OFFS = [00, 01, 10, 11]): 4-wide lat-lerp, 2-wide lon-lerp.
    float o[8];
    #pragma unroll
    for (int L = 0; L < 4; ++L) {
        float wla = wlat[L];
        float wlo = wlon[L];
        vf4 rl = row0[L] * splat4(1.0f - wla) + row1[L] * splat4(wla);
        vf2 f  = (vf2){rl.x, rl.y} * (1.0f - wlo) + (vf2){rl.z, rl.w} * wlo;
        o[2 * L + 0] = f.x;
        o[2 * L + 1] = f.y;
    }

    // Streaming 64 MB output: two NT b128 stores (32B/point, 16B aligned)
    // so the output stream doesn't evict the L2-resident embedding tables.
    vf4 lo = {o[0], o[1], o[2], o[3]};
    vf4 hi = {o[4], o[5], o[6], o[7]};
    vf4* ov = (vf4*)(out + (size_t)i * 8);
    __builtin_nontemporal_store(lo, ov + 0);
    __builtin_nontemporal_store(hi, ov + 1);
}

extern "C" void kernel_launch(void* const* d_in, const int* in_sizes, int n_in,
                              void* d_out, int out_size, void* d_ws, size_t ws_size,
                              hipStream_t stream) {
    const float* x   = (const float*)d_in[0];   // (B, 2) f32
    const float* emb = (const float*)d_in[1];   // (4, NGP_TABLE, 2) f32
    float* out       = (float*)d_out;           // (B, 8) f32

    int B = in_sizes[0] / 2;
    int blocks = (B + 255) / 256;
    ngp_interp_enc_kernel<<<blocks, 256, 0, stream>>>(x, emb, out, B);
}